// ExtractorMLP_42597485642493
// MI455X (gfx1250) — compile-verified
//
#include <hip/hip_runtime.h>

// ---------------- problem constants ----------------
#define E_EDGES   500000
#define NT_TILES  (E_EDGES / 16)       // 31250 exact
#define NG        512                  // graphs
#define HDIM      128
#define C1        512                  // 4*H
#define EPSV      1e-5f

// LDS padded row strides (words); %4==0 for float4 staging, conflict-free b64 frag reads.
#define APAD1  260
#define APAD2  516

typedef __attribute__((ext_vector_type(2))) float v2f;
typedef __attribute__((ext_vector_type(8))) float v8f;

// ---------------- sorting: hist / scan / scatter ----------------
__global__ __launch_bounds__(512) void k_init(int* __restrict__ hist) {
    hist[threadIdx.x] = 0;
}

__global__ __launch_bounds__(256) void k_hist(const int* __restrict__ eidx,
                                              const int* __restrict__ batch,
                                              int* __restrict__ segE,
                                              int* __restrict__ hist) {
    int e = blockIdx.x * 256 + threadIdx.x;
    if (e < E_EDGES) {
        int s = batch[eidx[e]];            // seg = batch[col[e]]
        segE[e] = s;
        atomicAdd(&hist[s], 1);
    }
}

__global__ __launch_bounds__(512) void k_scan(const int* __restrict__ hist,
                                              int* __restrict__ offsets,
                                              int* __restrict__ cur) {
    __shared__ int tmp[NG];
    int t = threadIdx.x;
    tmp[t] = hist[t];
    __syncthreads();
    for (int d = 1; d < NG; d <<= 1) {
        int v = (t >= d) ? tmp[t - d] : 0;
        __syncthreads();
        tmp[t] += v;
        __syncthreads();
    }
    int incl = tmp[t];
    int excl = incl - hist[t];
    offsets[t] = excl;
    cur[t]     = excl;
    if (t == NG - 1) offsets[NG] = incl;   // == E
}

__global__ __launch_bounds__(256) void k_scatter(const int* __restrict__ segE,
                                                 int* __restrict__ cur,
                                                 int* __restrict__ perm,
                                                 int* __restrict__ segS) {
    int e = blockIdx.x * 256 + threadIdx.x;
    if (e < E_EDGES) {
        int s = segE[e];
        int p = atomicAdd(&cur[s], 1);
        perm[p] = e;
        segS[p] = s;
    }
}

// ---------------- GEMM1: h1[pos,512] = cat(emb[col],emb[row]) @ W1 + b1 ----------------
// grid.x = 2 N-slabs of 256 cols; 8 waves, each owns TWO 16-col subtiles with
// register-resident B fragments (2 x 64 v2f = 256 VGPRs), A tile staged in LDS.
__global__ __launch_bounds__(256)
void k_gemm1(const float* __restrict__ emb, const int* __restrict__ eidx,
             const int* __restrict__ perm, const float* __restrict__ W1,
             const float* __restrict__ b1, float* __restrict__ h1, int numTiles) {
    __shared__ float sA[16 * APAD1];
    __shared__ int   sNode[32];               // col[0..15], row[16..31]

    const int tid  = threadIdx.x;
    const int nb   = blockIdx.x * 256;        // slab base column
    const int wave = tid >> 5, lane = tid & 31;
    const int half = lane >> 4, r = lane & 15;
    const int n0   = nb + wave * 32;          // first subtile base col (global)
    const int n1   = n0 + 16;                 // second subtile

    // load B fragments into registers (W1 is L2-resident, 512 KB)
    v2f breg0[64], breg1[64];
    #pragma unroll
    for (int kk = 0; kk < 64; ++kk) {
        int k = kk * 4 + half * 2;
        breg0[kk].x = W1[(k    ) * C1 + n0 + r];
        breg0[kk].y = W1[(k + 1) * C1 + n0 + r];
        breg1[kk].x = W1[(k    ) * C1 + n1 + r];
        breg1[kk].y = W1[(k + 1) * C1 + n1 + r];
    }
    const float bias0 = b1[n0 + r];
    const float bias1 = b1[n1 + r];

    for (int tile = blockIdx.y; tile < numTiles; tile += gridDim.y) {
        __syncthreads();                       // protects sA/sNode reuse
        if (tid < 16) {
            int e = perm[tile * 16 + tid];
            sNode[tid]      = eidx[e];             // col
            sNode[16 + tid] = eidx[E_EDGES + e];   // row
        }
        __syncthreads();
        // stage gathered A tile 16x256 (float4)
        for (int idx = tid; idx < 16 * 64; idx += 256) {
            int m = idx >> 6, k4 = (idx & 63) << 2;
            int node = (k4 < 128) ? sNode[m] : sNode[16 + m];
            *(float4*)(sA + m * APAD1 + k4) =
                *(const float4*)(emb + (size_t)node * HDIM + (k4 & 127));
        }
        __syncthreads();

        v8f acc0 = {}, acc1 = {};
        #pragma unroll
        for (int kk = 0; kk < 64; ++kk) {
            v2f a = *(const v2f*)(sA + r * APAD1 + kk * 4 + half * 2);
            acc0 = __builtin_amdgcn_wmma_f32_16x16x4_f32(false, a, false, breg0[kk],
                                                         (short)0, acc0, false, false);
            acc1 = __builtin_amdgcn_wmma_f32_16x16x4_f32(false, a, false, breg1[kk],
                                                         (short)0, acc1, false, false);
        }
        size_t rowBase = (size_t)(tile * 16) * C1 + n0 + r;
        #pragma unroll
        for (int v = 0; v < 8; ++v) {
            h1[rowBase + (size_t)(v + 8 * half) * C1]      = acc0[v] + bias0;
            h1[rowBase + (size_t)(v + 8 * half) * C1 + 16] = acc1[v] + bias1;
        }
    }
}

// ---------------- segment stats: mean / rsqrt(var+eps) per (graph, channel) ----------------
// grid = (NG, S/128); x is stored in sorted-edge order, row stride S.
__global__ __launch_bounds__(256)
void k_stats(const float* __restrict__ x, const int* __restrict__ offsets,
             float* __restrict__ mean, float* __restrict__ rsig, int S) {
    __shared__ float sS[256], sQ[256];
    const int g = blockIdx.x;
    const int cbase = blockIdx.y * 128;
    const int c  = threadIdx.x & 127;
    const int rr = threadIdx.x >> 7;
    const int start = offsets[g], end = offsets[g + 1];
    float s = 0.f, q = 0.f;
    for (int pos = start + rr; pos < end; pos += 2) {
        float v = x[(size_t)pos * S + cbase + c];
        s += v; q += v * v;
    }
    sS[threadIdx.x] = s; sQ[threadIdx.x] = q;
    __syncthreads();
    if (threadIdx.x < 128) {
        s = sS[threadIdx.x] + sS[threadIdx.x + 128];
        q = sQ[threadIdx.x] + sQ[threadIdx.x + 128];
        float cnt = (float)((end - start) > 0 ? (end - start) : 1);
        float m = s / cnt;
        float var = fmaxf(q / cnt - m * m, 0.f);
        mean[(size_t)g * S + cbase + c] = m;
        rsig[(size_t)g * S + cbase + c] = rsqrtf(var + EPSV);
    }
}

// ---------------- GEMM2: h2[pos,128] = relu(IN(h1)) @ W2 + b2 ----------------
// 8 waves, each owns one 16-col subtile with register-resident B (128 v2f = 256 VGPRs).
__global__ __launch_bounds__(256)
void k_gemm2(const float* __restrict__ h1, const float* __restrict__ mean1,
             const float* __restrict__ rsig1, const int* __restrict__ segS,
             const float* __restrict__ W2, const float* __restrict__ b2,
             float* __restrict__ h2, int numTiles) {
    __shared__ float sA[16 * APAD2];
    __shared__ int   sSeg[16];

    const int tid = threadIdx.x;
    const int wave = tid >> 5, lane = tid & 31;
    const int half = lane >> 4, r = lane & 15;
    const int n0 = wave * 16;

    // load B fragments into registers (W2 is L2-resident, 256 KB)
    v2f breg[128];
    #pragma unroll
    for (int kk = 0; kk < 128; ++kk) {
        int k = kk * 4 + half * 2;
        breg[kk].x = W2[(k    ) * HDIM + n0 + r];
        breg[kk].y = W2[(k + 1) * HDIM + n0 + r];
    }
    const float bias = b2[n0 + r];

    for (int tile = blockIdx.x; tile < numTiles; tile += gridDim.x) {
        __syncthreads();
        if (tid < 16) sSeg[tid] = segS[tile * 16 + tid];
        __syncthreads();
        // stage normalized+relu A tile 16x512 (float4)
        for (int idx = tid; idx < 16 * 128; idx += 256) {
            int m = idx >> 7, k4 = (idx & 127) << 2;
            int g = sSeg[m];
            float4 hv = *(const float4*)(h1 + (size_t)(tile * 16 + m) * C1 + k4);
            float4 mv = *(const float4*)(mean1 + (size_t)g * C1 + k4);
            float4 rv = *(const float4*)(rsig1 + (size_t)g * C1 + k4);
            float4 o;
            o.x = fmaxf((hv.x - mv.x) * rv.x, 0.f);
            o.y = fmaxf((hv.y - mv.y) * rv.y, 0.f);
            o.z = fmaxf((hv.z - mv.z) * rv.z, 0.f);
            o.w = fmaxf((hv.w - mv.w) * rv.w, 0.f);
            *(float4*)(sA + m * APAD2 + k4) = o;
        }
        __syncthreads();

        v8f acc = {};
        #pragma unroll
        for (int kk = 0; kk < 128; ++kk) {
            v2f a = *(const v2f*)(sA + r * APAD2 + kk * 4 + half * 2);
            acc = __builtin_amdgcn_wmma_f32_16x16x4_f32(false, a, false, breg[kk],
                                                        (short)0, acc, false, false);
        }
        size_t rowBase = (size_t)(tile * 16) * HDIM + n0 + r;
        #pragma unroll
        for (int v = 0; v < 8; ++v)
            h2[rowBase + (size_t)(v + 8 * half) * HDIM] = acc[v] + bias;
    }
}

// ---------------- final: out[perm[pos]] = relu(IN(h2)) . W3 + b3 ----------------
__global__ __launch_bounds__(256)
void k_final(const float* __restrict__ h2, const float* __restrict__ mean2,
             const float* __restrict__ rsig2, const int* __restrict__ segS,
             const int* __restrict__ perm, const float* __restrict__ W3,
             const float* __restrict__ b3, float* __restrict__ out) {
    int pos  = blockIdx.x * 8 + (threadIdx.x >> 5);
    int lane = threadIdx.x & 31;
    if (pos >= E_EDGES) return;
    int g = segS[pos];
    const float* hrow = h2    + (size_t)pos * HDIM;
    const float* mrow = mean2 + (size_t)g * HDIM;
    const float* rrow = rsig2 + (size_t)g * HDIM;
    float acc = 0.f;
    #pragma unroll
    for (int j = 0; j < 4; ++j) {
        int c = lane + 32 * j;
        float v = fmaxf((hrow[c] - mrow[c]) * rrow[c], 0.f);
        acc += v * W3[c];
    }
    #pragma unroll
    for (int off = 16; off > 0; off >>= 1) acc += __shfl_down(acc, off, 32);
    if (lane == 0) out[perm[pos]] = acc + b3[0];
}

// ---------------- launcher ----------------
extern "C" void kernel_launch(void* const* d_in, const int* in_sizes, int n_in,
                              void* d_out, int out_size, void* d_ws, size_t ws_size,
                              hipStream_t stream) {
    const float* emb   = (const float*)d_in[0];
    const int*   eidx  = (const int*)  d_in[1];
    const int*   batch = (const int*)  d_in[2];
    const float* W1    = (const float*)d_in[3];
    const float* b1    = (const float*)d_in[4];
    const float* W2    = (const float*)d_in[5];
    const float* b2    = (const float*)d_in[6];
    const float* W3    = (const float*)d_in[7];
    const float* b3    = (const float*)d_in[8];
    float* out = (float*)d_out;

    char* ws = (char*)d_ws;
    auto carve = [&](size_t bytes) -> char* {
        char* p = ws;
        ws += (bytes + 255) & ~(size_t)255;
        return p;
    };
    float* h1      = (float*)carve((size_t)E_EDGES * C1   * sizeof(float));
    float* h2      = (float*)carve((size_t)E_EDGES * HDIM * sizeof(float));
    int*   segE    = (int*)  carve((size_t)E_EDGES * sizeof(int));
    int*   perm    = (int*)  carve((size_t)E_EDGES * sizeof(int));
    int*   segS    = (int*)  carve((size_t)E_EDGES * sizeof(int));
    int*   hist    = (int*)  carve(NG * sizeof(int));
    int*   offs    = (int*)  carve((NG + 1) * sizeof(int));
    int*   cur     = (int*)  carve(NG * sizeof(int));
    float* mean1   = (float*)carve((size_t)NG * C1   * sizeof(float));
    float* rsig1   = (float*)carve((size_t)NG * C1   * sizeof(float));
    float* mean2   = (float*)carve((size_t)NG * HDIM * sizeof(float));
    float* rsig2   = (float*)carve((size_t)NG * HDIM * sizeof(float));

    const int EB = (E_EDGES + 255) / 256;

    k_init   <<<1, 512, 0, stream>>>(hist);
    k_hist   <<<EB, 256, 0, stream>>>(eidx, batch, segE, hist);
    k_scan   <<<1, 512, 0, stream>>>(hist, offs, cur);
    k_scatter<<<EB, 256, 0, stream>>>(segE, cur, perm, segS);

    k_gemm1  <<<dim3(2, 1024), 256, 0, stream>>>(emb, eidx, perm, W1, b1, h1, NT_TILES);
    k_stats  <<<dim3(NG, 4), 256, 0, stream>>>(h1, offs, mean1, rsig1, C1);
    k_gemm2  <<<dim3(1024), 256, 0, stream>>>(h1, mean1, rsig1, segS, W2, b2, h2, NT_TILES);
    k_stats  <<<dim3(NG, 1), 256, 0, stream>>>(h2, offs, mean2, rsig2, HDIM);
    k_final  <<<(E_EDGES + 7) / 8, 256, 0, stream>>>(h2, mean2, rsig2, segS, perm, W3, b3, out);
}